// Attention_9689446220182
// MI455X (gfx1250) — compile-verified
//
#include <hip/hip_runtime.h>

typedef __attribute__((ext_vector_type(2))) float v2f;
typedef __attribute__((ext_vector_type(8))) float v8f;

#define WAVES 8
#define TK 16           // keys per wave per tile
#define DD 128
#define LK 4096
#define LQ 4
#define NBH 256         // B*H = 8*32
#define HH 32
#define SCALE_F 0.08838834764831845f

__global__ __launch_bounds__(256) void attn_fa_kernel(
    const float* __restrict__ q, const float* __restrict__ k,
    const float* __restrict__ v, const float* __restrict__ mask,
    float* __restrict__ out)
{
    __shared__ float pbuf[WAVES][LQ][TK];    // P transpose bounce (C layout -> A layout)
    __shared__ float sO[WAVES][LQ][DD];      // per-wave partial outputs
    __shared__ float sM[WAVES][LQ];          // per-wave row maxes
    __shared__ float sL[WAVES][LQ];          // per-wave row sums

    const int bh     = blockIdx.x;
    const int b      = bh / HH;
    const int tid    = threadIdx.x;
    const int wave   = tid >> 5;
    const int lane   = tid & 31;
    const int laneHi = lane >> 4;            // which half of the wave
    const int lane16 = lane & 15;
    const int rowA   = lane16 & 3;           // clamped query row for A-operand (avoids OOB)

    const float* qp = q    + (size_t)bh * LQ * DD;
    const float* kp = k    + (size_t)bh * LK * DD;
    const float* vp = v    + (size_t)bh * LK * DD;
    const float* mp = mask + (size_t)b  * LQ * LK;

    // ---- Q A-operand: 32 slices of K=4. Per ISA 32-bit A 16x4 layout:
    // lane L holds M=L%16; VGPR j holds K = 2*(L/16)+j  ->  float2 at d = 4*s + 2*laneHi.
    // Q is reused every tile by this wave: keep regular (temporal) loads.
    v2f qa[32];
#pragma unroll
    for (int s = 0; s < 32; ++s) {
        const float* src = qp + rowA * DD + 4 * s + 2 * laneHi;
        qa[s] = (v2f){src[0], src[1]};
    }

    // Output accumulators: 8 C tiles of 16x16 (d = 16*t + n)
    v8f oacc[8];
#pragma unroll
    for (int t = 0; t < 8; ++t) oacc[t] = (v8f){0.f,0.f,0.f,0.f,0.f,0.f,0.f,0.f};

    float mrow[4] = {-3.0e38f, -3.0e38f, -3.0e38f, -3.0e38f};
    float lrow[4] = {0.f, 0.f, 0.f, 0.f};

    const int kbase0 = wave * (LK / WAVES);  // 512 keys per wave

    for (int kt = 0; kt < LK / WAVES; kt += TK) {
        const int keybase = kbase0 + kt;
        const int myKey   = keybase + lane16;

        // ---- scores tile: C[16x16] = Q[16x128] * K^T[128x16]
        // B layout: lane L holds N=L%16 (key); VGPR j holds K-dim = 2*laneHi + j.
        // K is streamed once (no reuse anywhere): non-temporal to spare L2/WGP$.
        v8f c = (v8f){0.f,0.f,0.f,0.f,0.f,0.f,0.f,0.f};
        const float* krow = kp + (size_t)myKey * DD + 2 * laneHi;
#pragma unroll
        for (int s = 0; s < 32; ++s) {
            v2f bb = __builtin_nontemporal_load((const v2f*)(krow + 4 * s));
            c = __builtin_amdgcn_wmma_f32_16x16x4_f32(
                    false, qa[s], false, bb, (short)0, c, false, false);
        }

        // ---- online softmax on C layout (VGPR r = row r; lanes 16-31 duplicate row r)
#pragma unroll
        for (int r = 0; r < 4; ++r) {
            // mask is broadcast over H (32 WGs share each row): keep it temporal.
            float sc = c[r] * SCALE_F + mp[r * LK + keybase + lane16];

            float rm = sc;                                   // row max over lanes 0-15
            rm = fmaxf(rm, __shfl_xor(rm, 8, 32));
            rm = fmaxf(rm, __shfl_xor(rm, 4, 32));
            rm = fmaxf(rm, __shfl_xor(rm, 2, 32));
            rm = fmaxf(rm, __shfl_xor(rm, 1, 32));
            rm = __shfl(rm, 0, 32);                          // broadcast

            float mnew  = fmaxf(mrow[r], rm);
            float alpha = __expf(mrow[r] - mnew);
            mrow[r] = mnew;

            float p = __expf(sc - mnew);

            float rs = p;                                    // row sum over lanes 0-15
            rs += __shfl_xor(rs, 8, 32);
            rs += __shfl_xor(rs, 4, 32);
            rs += __shfl_xor(rs, 2, 32);
            rs += __shfl_xor(rs, 1, 32);
            rs = __shfl(rs, 0, 32);

            lrow[r] = lrow[r] * alpha + rs;

#pragma unroll
            for (int t = 0; t < 8; ++t) oacc[t][r] *= alpha; // rescale accumulators

            if (lane < 16) pbuf[wave][r][lane16] = p;        // stage for A-layout read
        }

        // ---- PV: oacc[t] += P[16x16] * V[16x128], K=4 steps over the 16 keys
#pragma unroll
        for (int ks = 0; ks < 4; ++ks) {
            v2f pa;
            pa.x = pbuf[wave][rowA][4 * ks + 2 * laneHi + 0];
            pa.y = pbuf[wave][rowA][4 * ks + 2 * laneHi + 1];
            const int keyk = keybase + 4 * ks + 2 * laneHi;
#pragma unroll
            for (int t = 0; t < 8; ++t) {
                // V streamed once: non-temporal.
                const float* vr = vp + (size_t)keyk * DD + 16 * t + lane16;
                v2f vb;
                vb.x = __builtin_nontemporal_load(vr);
                vb.y = __builtin_nontemporal_load(vr + DD);
                oacc[t] = __builtin_amdgcn_wmma_f32_16x16x4_f32(
                              false, pa, false, vb, (short)0, oacc[t], false, false);
            }
        }
    }

    // ---- split-K merge across the 8 waves via LDS (deterministic, no FP atomics)
#pragma unroll
    for (int r = 0; r < 4; ++r) {
        if (lane < 16) {
#pragma unroll
            for (int t = 0; t < 8; ++t) sO[wave][r][16 * t + lane16] = oacc[t][r];
        }
        if (lane == 0) { sM[wave][r] = mrow[r]; sL[wave][r] = lrow[r]; }
    }
    __syncthreads();

    for (int idx = tid; idx < LQ * DD; idx += 256) {
        const int r = idx / DD;
        const int d = idx % DD;
        float M = -3.0e38f;
        for (int w = 0; w < WAVES; ++w) M = fmaxf(M, sM[w][r]);
        float L = 0.f, O = 0.f;
        for (int w = 0; w < WAVES; ++w) {
            float e = __expf(sM[w][r] - M);
            L += sL[w][r] * e;
            O += sO[w][r][d] * e;
        }
        out[(size_t)bh * LQ * DD + idx] = O / L;
    }
}

extern "C" void kernel_launch(void* const* d_in, const int* in_sizes, int n_in,
                              void* d_out, int out_size, void* d_ws, size_t ws_size,
                              hipStream_t stream) {
    const float* q    = (const float*)d_in[0];
    const float* k    = (const float*)d_in[1];
    const float* v    = (const float*)d_in[2];
    const float* mask = (const float*)d_in[3];
    float* out = (float*)d_out;
    (void)in_sizes; (void)n_in; (void)out_size; (void)d_ws; (void)ws_size;
    hipLaunchKernelGGL(attn_fa_kernel, dim3(NBH), dim3(256), 0, stream,
                       q, k, v, mask, out);
}